// DigitCapsules_77077483094602
// MI455X (gfx1250) — compile-verified
//
#include <hip/hip_runtime.h>

typedef float v2f __attribute__((ext_vector_type(2)));
typedef float v8f __attribute__((ext_vector_type(8)));

#define BATCH 256
#define N1 1152
#define N2 10
#define D2 16
#define D1 8
#define EPS 1e-8f

// ---------------------------------------------------------------------------
// Kernel 1: u_hat[b,i,j,d] = sum_k x[b,i,k] * W[i,j,d,k]
// Per i: GEMM  [B=256 x K=8] @ [K=8 x (j*d=160)].  One 16(batch) x 16(d) tile
// for one output capsule j == two chained V_WMMA_F32_16X16X4_F32 (K = 2*4).
// grid = N1; block = 320 threads = 10 waves, wave w -> j = w.  Each wave keeps
// its B operand (W[i,j]) in registers and loops all 16 batch tiles -> 32 WMMAs
// per wave, W loaded exactly once.
// ---------------------------------------------------------------------------
__global__ __launch_bounds__(320)
void uhat_wmma_kernel(const float* __restrict__ x, const float* __restrict__ W,
                      float* __restrict__ uhat) {
  const int i    = blockIdx.x;
  const int j    = threadIdx.x >> 5;          // wave id = output capsule
  const int lane = threadIdx.x & 31;
  const int half = lane >> 4;                 // 0: lanes 0-15, 1: lanes 16-31
  const int l    = lane & 15;

  // B operand = W[i,j]^T tile [4 k x 16 d]; N = lane, half-lane K split
  // (VGPR0: K=0 | K=2, VGPR1: K=1 | K=3).  Loaded once, reused for all tiles.
  const float* wp = W + (((size_t)i * N2 + j) * D2 + l) * D1;
  v2f b0, b1;
  b0.x = wp[0 + 2 * half]; b0.y = wp[1 + 2 * half];   // K = 0..3
  b1.x = wp[4 + 2 * half]; b1.y = wp[5 + 2 * half];   // K = 4..7

  const float* xcol  = x + (size_t)i * D1;
  float*       ubase = uhat + ((size_t)i * N2 + j) * D2 + l;

#pragma unroll 2
  for (int mt = 0; mt < BATCH / 16; ++mt) {
    const int mbase = mt << 4;
    // A operand = x tile [16 batch x 4 k], M = l on both lane halves.
    const float* xp = xcol + (size_t)(mbase + l) * (N1 * D1);
    v2f a0, a1;
    a0.x = xp[0 + 2 * half]; a0.y = xp[1 + 2 * half];
    a1.x = xp[4 + 2 * half]; a1.y = xp[5 + 2 * half];

    v8f c = {};
    // 8 args: (neg_a, A, neg_b, B, c_mod, C, reuse_a, reuse_b)
    c = __builtin_amdgcn_wmma_f32_16x16x4_f32(false, a0, false, b0, (short)0, c, false, false);
    c = __builtin_amdgcn_wmma_f32_16x16x4_f32(false, a1, false, b1, (short)0, c, false, false);

    // C/D layout: VGPR r, lanes 0-15 -> M=r, lanes 16-31 -> M=r+8; N = lane.
    // Each store: 16 lanes write 16 consecutive d -> 64B contiguous segments.
    float* up = ubase + (size_t)(mbase + 8 * half) * (N1 * N2 * D2);
#pragma unroll
    for (int r = 0; r < 8; ++r)
      up[(size_t)r * (N1 * N2 * D2)] = c[r];
  }
}

// ---------------------------------------------------------------------------
// Kernel 2: fused 3-iteration dynamic routing, one workgroup per batch elem.
// b-logits [1152 x 10] live in LDS.  Threads = (isub 0..15, d 0..15).
// u_hat (189 MB) is expected to stay resident in the 192 MB global L2.
// ---------------------------------------------------------------------------
__global__ __launch_bounds__(256)
void routing_kernel(const float* __restrict__ uhat, float* __restrict__ out) {
  const int b    = blockIdx.x;
  const int tid  = threadIdx.x;
  const int d    = tid & 15;
  const int isub = tid >> 4;                  // 0..15

  __shared__ float lds_b[N1 * N2];            // 46080 B routing logits
  __shared__ float lds_part[16][N2][D2];      // 10240 B partial s
  __shared__ float lds_s[N2][D2];
  __shared__ float lds_scale[N2];
  __shared__ float lds_v[N2][D2];

  for (int t = tid; t < N1 * N2; t += 256) lds_b[t] = 0.f;
  __syncthreads();

  const float* ub = uhat + (size_t)b * (N1 * N2 * D2);

  for (int it = 0; it < 3; ++it) {
    // ---- phase 1: c = softmax(b) over j; s partial-accumulate over i ----
    float acc[N2];
#pragma unroll
    for (int j = 0; j < N2; ++j) acc[j] = 0.f;

    for (int i = isub; i < N1; i += 16) {
      const float* up = ub + (size_t)i * (N2 * D2) + d;
      __builtin_prefetch(up + 16 * N2 * D2, 0, 1);    // next i row of u_hat
      float br[N2];
      float m = -1e30f;
#pragma unroll
      for (int j = 0; j < N2; ++j) { br[j] = lds_b[i * N2 + j]; m = fmaxf(m, br[j]); }
      float sum = 0.f;
#pragma unroll
      for (int j = 0; j < N2; ++j) { br[j] = __expf(br[j] - m); sum += br[j]; }
      const float inv = 1.f / sum;
#pragma unroll
      for (int j = 0; j < N2; ++j) acc[j] += (br[j] * inv) * up[j * D2];
    }
#pragma unroll
    for (int j = 0; j < N2; ++j) lds_part[isub][j][d] = acc[j];
    __syncthreads();

    // ---- phase 2: finish s reduction, squash into v ----
    if (tid < N2 * D2) {
      const int j = tid >> 4, dd = tid & 15;
      float s = 0.f;
#pragma unroll
      for (int g = 0; g < 16; ++g) s += lds_part[g][j][dd];
      lds_s[j][dd] = s;
    }
    __syncthreads();
    if (tid < N2) {
      float n2 = 0.f;
#pragma unroll
      for (int dd = 0; dd < D2; ++dd) { const float sv = lds_s[tid][dd]; n2 += sv * sv; }
      lds_scale[tid] = (n2 / (1.f + n2)) * __frsqrt_rn(n2 + EPS);
    }
    __syncthreads();
    if (tid < N2 * D2) {
      const int j = tid >> 4, dd = tid & 15;
      lds_v[j][dd] = lds_s[j][dd] * lds_scale[j];
    }
    __syncthreads();

    // ---- phase 3: agreement update  b[i,j] += <u_hat[i,j,:], v[j,:]> ----
    if (it < 2) {
      for (int i = isub; i < N1; i += 16) {
        const float* up = ub + (size_t)i * (N2 * D2) + d;
#pragma unroll
        for (int j = 0; j < N2; ++j) {
          float p = up[j * D2] * lds_v[j][d];
          // reduce across the 16 d-lanes (wave32, width-16 subgroups)
          p += __shfl_xor(p, 1, 16);
          p += __shfl_xor(p, 2, 16);
          p += __shfl_xor(p, 4, 16);
          p += __shfl_xor(p, 8, 16);
          if (d == 0) lds_b[i * N2 + j] += p;   // unique owner per (i,j)
        }
      }
      __syncthreads();
    }
  }

  if (tid < N2 * D2) out[(size_t)b * (N2 * D2) + tid] = lds_v[tid >> 4][tid & 15];
}

// ---------------------------------------------------------------------------
extern "C" void kernel_launch(void* const* d_in, const int* in_sizes, int n_in,
                              void* d_out, int out_size, void* d_ws, size_t ws_size,
                              hipStream_t stream) {
  const float* x = (const float*)d_in[0];   // [256, 1152, 8]
  const float* W = (const float*)d_in[1];   // [1152, 10, 16, 8]
  float* out  = (float*)d_out;              // [256, 10, 16]
  float* uhat = (float*)d_ws;               // [256, 1152, 10, 16] = 189 MB

  uhat_wmma_kernel<<<N1, 320, 0, stream>>>(x, W, uhat);
  routing_kernel<<<BATCH, 256, 0, stream>>>(uhat, out);
}